// GATConv_69002944577595
// MI455X (gfx1250) — compile-verified
//
#include <hip/hip_runtime.h>
#include <hip/hip_bf16.h>
#include <math.h>

// Problem constants (from reference)
#define N_NODES 8192
#define F_IN    256
#define O_OUT   64
#define ALPHA   0.02f

typedef __attribute__((ext_vector_type(2))) float v2f;
typedef __attribute__((ext_vector_type(8))) float v8f;

// ---------------------------------------------------------------------------
// Workspace layout (float-offsets unless noted):
//   h   : [0,        524288)    8192x64 f32 = x @ W1
//   s   : 524288  (+8192)       h @ a_src
//   d   : 532480  (+8192)       h @ a_dst
//   b1  : 540672  (+8192)       exp(d_j)
//   b2  : 548864  (+8192)       exp(ALPHA*d_j)
//   a1  : 557056  (+8192)       exp(s_i - m_i)/Z_i
//   a2  : 565248  (+8192)       exp(ALPHA*s_i - m_i)/Z_i
//   hp  : 573440  (+4*524288)   4 K-segment partial buffers of h_prime
//   pk  : u32 @ float-off 2670592: 8192x256 packed adjacency bits (8MB)
// Total ~19.1 MB.
// ---------------------------------------------------------------------------
#define WS_H   0
#define WS_S   524288
#define WS_D   532480
#define WS_B1  540672
#define WS_B2  548864
#define WS_A1  557056
#define WS_A2  565248
#define WS_HP  573440
#define WS_PK  2670592          // float-offset; reinterpret as uint32*
#define HP_SEG 524288           // per-segment stride (floats)

// ---------------------------------------------------------------------------
// Kernel 1: h = x @ W1 using V_WMMA_F32_16X16X4_F32.
// One wave computes a 16-row x 64-col tile of h. 8 waves per block.
// A (16x4 f32): lanes 0-15 hold M=0..15 / K=k,k+1 ; lanes 16-31 K=k+2,k+3.
// B (4x16 f32): lanes 0-15 hold K=k,k+1 ; lanes 16-31 K=k+2,k+3 ; N=lane&15.
// C (16x16 f32): VGPR v -> row v (lanes 0-15) / row v+8 (lanes 16-31).
// ---------------------------------------------------------------------------
__global__ void k1_gemm_h(const float* __restrict__ x,
                          const float* __restrict__ W1,
                          float* __restrict__ h) {
    const int wave = threadIdx.x >> 5;
    const int lane = threadIdx.x & 31;
    const int half = lane >> 4;
    const int r    = lane & 15;
    const int m0   = (blockIdx.x * 8 + wave) * 16;

    v8f c0 = {}, c1 = {}, c2 = {}, c3 = {};
    const int kh = half ? 2 : 0;

    for (int k = 0; k < F_IN; k += 4) {
        v2f a;
        a.x = x[(size_t)(m0 + r) * F_IN + k + kh];
        a.y = x[(size_t)(m0 + r) * F_IN + k + kh + 1];
        v2f b;
        b.x = W1[(size_t)(k + kh) * O_OUT + 0 * 16 + r];
        b.y = W1[(size_t)(k + kh + 1) * O_OUT + 0 * 16 + r];
        c0 = __builtin_amdgcn_wmma_f32_16x16x4_f32(false, a, false, b, (short)0, c0, false, false);
        b.x = W1[(size_t)(k + kh) * O_OUT + 1 * 16 + r];
        b.y = W1[(size_t)(k + kh + 1) * O_OUT + 1 * 16 + r];
        c1 = __builtin_amdgcn_wmma_f32_16x16x4_f32(false, a, false, b, (short)0, c1, false, false);
        b.x = W1[(size_t)(k + kh) * O_OUT + 2 * 16 + r];
        b.y = W1[(size_t)(k + kh + 1) * O_OUT + 2 * 16 + r];
        c2 = __builtin_amdgcn_wmma_f32_16x16x4_f32(false, a, false, b, (short)0, c2, false, false);
        b.x = W1[(size_t)(k + kh) * O_OUT + 3 * 16 + r];
        b.y = W1[(size_t)(k + kh + 1) * O_OUT + 3 * 16 + r];
        c3 = __builtin_amdgcn_wmma_f32_16x16x4_f32(false, a, false, b, (short)0, c3, false, false);
    }

    const int rowoff = half ? 8 : 0;
    #pragma unroll
    for (int v = 0; v < 8; ++v) {
        size_t row = (size_t)(m0 + v + rowoff) * O_OUT;
        h[row + 0 * 16 + r] = c0[v];
        h[row + 1 * 16 + r] = c1[v];
        h[row + 2 * 16 + r] = c2[v];
        h[row + 3 * 16 + r] = c3[v];
    }
}

// ---------------------------------------------------------------------------
// Kernel 2: s = h@a_src, d = h@a_dst, column factors b1=exp(d), b2=exp(a*d)
// ---------------------------------------------------------------------------
__global__ void k2_sd(const float* __restrict__ h,
                      const float* __restrict__ att,
                      float* __restrict__ s, float* __restrict__ d,
                      float* __restrict__ b1, float* __restrict__ b2) {
    int i = blockIdx.x * blockDim.x + threadIdx.x;
    float ss = 0.f, dd = 0.f;
    #pragma unroll 8
    for (int f = 0; f < O_OUT; ++f) {
        float hv = h[(size_t)i * O_OUT + f];
        ss = fmaf(hv, att[f], ss);
        dd = fmaf(hv, att[O_OUT + f], dd);
    }
    s[i] = ss;
    d[i] = dd;
    b1[i] = __expf(dd);
    b2[i] = __expf(ALPHA * dd);
}

// ---------------------------------------------------------------------------
// Kernel 3: one block per row i. Single streaming b128 read of adj row:
//  - bit-pack (nibble per lane, OR-combined over groups of 8 lanes with
//    3x shfl_xor) into LDS + global: adj read ONCE (268MB -> 8MB packed)
//  - dmax = max d_j over allowed j  =>  m_i = lrelu(s_i + dmax)  (monotone)
//  - second sweep over packed bits: Z_i via factored exp terms (no exp in loop)
//  - a1_i = exp(s_i-m)/Z ; a2_i = exp(ALPHA*s_i-m)/Z
// (A fully-masked row would need uniform attention; prob 2^-8192, ignored.)
// ---------------------------------------------------------------------------
__global__ void k3_rowstats(const int* __restrict__ adj,
                            const float* __restrict__ s,
                            const float* __restrict__ dv,
                            const float* __restrict__ b1,
                            const float* __restrict__ b2,
                            unsigned int* __restrict__ packed,
                            float* __restrict__ a1, float* __restrict__ a2) {
    __shared__ unsigned int pk[256];   // 256 words = 8192 bits (one row)
    __shared__ float red[256];

    const int i    = blockIdx.x;
    const int tid  = threadIdx.x;
    const int lane = tid & 31;
    const size_t rowbase = (size_t)i * N_NODES;

    const float si = s[i];

    float lmax = -3.0e38f;
    #pragma unroll 2
    for (int k = 0; k < 8; ++k) {
        const int j0 = k * 1024 + tid * 4;           // 4 consecutive j per lane
        if (k + 1 < 8)
            __builtin_prefetch(&adj[rowbase + j0 + 1024], 0, 1);
        int4   av = *reinterpret_cast<const int4*>(&adj[rowbase + j0]);
        float4 dj = *reinterpret_cast<const float4*>(&dv[j0]);

        unsigned nib = (av.x > 0 ? 1u : 0u) | (av.y > 0 ? 2u : 0u) |
                       (av.z > 0 ? 4u : 0u) | (av.w > 0 ? 8u : 0u);
        if (av.x > 0) lmax = fmaxf(lmax, dj.x);
        if (av.y > 0) lmax = fmaxf(lmax, dj.y);
        if (av.z > 0) lmax = fmaxf(lmax, dj.z);
        if (av.w > 0) lmax = fmaxf(lmax, dj.w);

        // OR-combine nibbles of 8 consecutive lanes into one 32-bit word
        unsigned wbits = nib << ((lane & 7) * 4);
        wbits |= __shfl_xor(wbits, 1);
        wbits |= __shfl_xor(wbits, 2);
        wbits |= __shfl_xor(wbits, 4);
        if ((lane & 7) == 0) {
            int widx = j0 >> 5;                      // word index within row
            pk[widx] = wbits;
            packed[(size_t)i * 256 + widx] = wbits;
        }
    }

    // block max-reduce
    red[tid] = lmax;
    __syncthreads();
    for (int st = 128; st > 0; st >>= 1) {
        if (tid < st) red[tid] = fmaxf(red[tid], red[tid + st]);
        __syncthreads();
    }
    const float dmax = red[0];
    __syncthreads();

    const float t_m = si + dmax;
    const float m   = (t_m >= 0.f) ? t_m : ALPHA * t_m;
    const float ea1 = __expf(si - m);
    const float ea2 = __expf(ALPHA * si - m);

    float z = 0.f;
    #pragma unroll 2
    for (int k = 0; k < 8; ++k) {
        const int j0 = k * 1024 + tid * 4;
        unsigned nib = (pk[j0 >> 5] >> ((tid & 7) * 4)) & 0xFu;
        float4 dj  = *reinterpret_cast<const float4*>(&dv[j0]);
        float4 e1  = *reinterpret_cast<const float4*>(&b1[j0]);
        float4 e2  = *reinterpret_cast<const float4*>(&b2[j0]);
        if (nib & 1u) { float t = si + dj.x; z += (t >= 0.f) ? ea1 * e1.x : ea2 * e2.x; }
        if (nib & 2u) { float t = si + dj.y; z += (t >= 0.f) ? ea1 * e1.y : ea2 * e2.y; }
        if (nib & 4u) { float t = si + dj.z; z += (t >= 0.f) ? ea1 * e1.z : ea2 * e2.z; }
        if (nib & 8u) { float t = si + dj.w; z += (t >= 0.f) ? ea1 * e1.w : ea2 * e2.w; }
    }
    red[tid] = z;
    __syncthreads();
    for (int st = 128; st > 0; st >>= 1) {
        if (tid < st) red[tid] += red[tid + st];
        __syncthreads();
    }
    if (tid == 0) {
        float inv = 1.0f / red[0];
        a1[i] = ea1 * inv;
        a2[i] = ea2 * inv;
    }
}

// ---------------------------------------------------------------------------
// Kernel 4: h_prime(seg) = attention[seg rows]^T @ h via V_WMMA_F32_16X16X4_F32.
// Attention weights generated in-register:
//   w(i,j) = adjbit(i,j) * ( (s_i+d_j >= 0) ? a1_i*b1_j : a2_i*b2_j )
// grid.x = 64 j-blocks of 128 columns; grid.y = 4 K-segments (2048 i each).
// 8 waves/block; wave handles 16 j-rows x all 64 features (4 C tiles).
// h chunk (64 i x 64 f) staged in LDS via GLOBAL_LOAD_ASYNC_TO_LDS_B128
// (ASYNCcnt path), stride 72 dwords so the two wave halves hit disjoint banks.
// Each segment writes its own partial buffer (deterministic; no atomics).
// ---------------------------------------------------------------------------
#define HS_STRIDE 72
__global__ void k4_attn_gemm(const unsigned int* __restrict__ packed,
                             const float* __restrict__ h,
                             const float* __restrict__ s,
                             const float* __restrict__ dv,
                             const float* __restrict__ b1,
                             const float* __restrict__ b2,
                             const float* __restrict__ a1,
                             const float* __restrict__ a2,
                             float* __restrict__ hp) {
    __shared__ float hs[64 * HS_STRIDE];   // 18KB staged h chunk
    __shared__ float sv[64], a1v[64], a2v[64];
    __shared__ unsigned int aw[64 * 4];    // adj bits: 64 i-rows x 128 j bits

    const int tid  = threadIdx.x;
    const int wave = tid >> 5;
    const int lane = tid & 31;
    const int half = lane >> 4;
    const int r    = lane & 15;

    const int jblock = blockIdx.x;              // 0..63 (128 j each)
    const int jlocal = wave * 16 + r;           // 0..127
    const int j      = jblock * 128 + jlocal;
    const int seg    = blockIdx.y;              // 0..3
    const int iseg0  = seg * 2048;

    const float djr = dv[j];
    const float b1j = b1[j];
    const float b2j = b2[j];

    v8f c0 = {}, c1 = {}, c2 = {}, c3 = {};

    const int srow = tid >> 2;      // 0..63  (staging row)
    const int sq   = tid & 3;       // 0..3   (16-float quarter / adj word)

    // LDS byte address for this thread's staging slot (aperture: addr[31:0])
    const unsigned ldsa =
        (unsigned)(uintptr_t)(&hs[srow * HS_STRIDE + sq * 16]);

    for (int chunk = 0; chunk < 32; ++chunk) {
        const int ibase = iseg0 + chunk * 64;

        // async-stage h[ibase+srow][sq*16 .. +15] : 4 x b128, offset applies
        // to both global and LDS sides (ISA 10.x async pseudocode).
        {
            const float* gsrc = &h[(size_t)(ibase + srow) * O_OUT + sq * 16];
            asm volatile(
                "global_load_async_to_lds_b128 %0, %1, off\n\t"
                "global_load_async_to_lds_b128 %0, %1, off offset:16\n\t"
                "global_load_async_to_lds_b128 %0, %1, off offset:32\n\t"
                "global_load_async_to_lds_b128 %0, %1, off offset:48"
                :: "v"(ldsa), "v"(gsrc)
                : "memory");
        }
        if (tid < 64) {
            sv[tid]  = s[ibase + tid];
            a1v[tid] = a1[ibase + tid];
            a2v[tid] = a2[ibase + tid];
        }
        aw[srow * 4 + sq] = packed[(size_t)(ibase + srow) * 256 + jblock * 4 + sq];
        asm volatile("s_wait_asynccnt 0x0" ::: "memory");
        __syncthreads();

        #pragma unroll 4
        for (int step = 0; step < 16; ++step) {
            const int iA = step * 4 + (half ? 2 : 0);   // K slot for VGPR0
            const int iB = iA + 1;                      // K slot for VGPR1

            unsigned wA = aw[iA * 4 + (jlocal >> 5)];
            unsigned wB = aw[iB * 4 + (jlocal >> 5)];
            float tA = sv[iA] + djr, tB = sv[iB] + djr;
            float pA = (tA >= 0.f) ? a1v[iA] * b1j : a2v[iA] * b2j;
            float pB = (tB >= 0.f) ? a1v[iB] * b1j : a2v[iB] * b2j;
            v2f a;
            a.x = ((wA >> (jlocal & 31)) & 1u) ? pA : 0.f;
            a.y = ((wB >> (jlocal & 31)) & 1u) ? pB : 0.f;

            v2f b;
            b.x = hs[iA * HS_STRIDE + 0 * 16 + r];
            b.y = hs[iB * HS_STRIDE + 0 * 16 + r];
            c0 = __builtin_amdgcn_wmma_f32_16x16x4_f32(false, a, false, b, (short)0, c0, false, false);
            b.x = hs[iA * HS_STRIDE + 1 * 16 + r];
            b.y = hs[iB * HS_STRIDE + 1 * 16 + r];
            c1 = __builtin_amdgcn_wmma_f32_16x16x4_f32(false, a, false, b, (short)0, c1, false, false);
            b.x = hs[iA * HS_STRIDE + 2 * 16 + r];
            b.y = hs[iB * HS_STRIDE + 2 * 16 + r];
            c2 = __builtin_amdgcn_wmma_f32_16x16x4_f32(false, a, false, b, (short)0, c2, false, false);
            b.x = hs[iA * HS_STRIDE + 3 * 16 + r];
            b.y = hs[iB * HS_STRIDE + 3 * 16 + r];
            c3 = __builtin_amdgcn_wmma_f32_16x16x4_f32(false, a, false, b, (short)0, c3, false, false);
        }
        __syncthreads();
    }

    // write partial tile to this segment's buffer (disjoint; deterministic)
    float* hps = hp + (size_t)seg * HP_SEG;
    const int rowoff = half ? 8 : 0;
    #pragma unroll
    for (int v = 0; v < 8; ++v) {
        size_t row = (size_t)(jblock * 128 + wave * 16 + v + rowoff) * O_OUT;
        hps[row + 0 * 16 + r] = c0[v];
        hps[row + 1 * 16 + r] = c1[v];
        hps[row + 2 * 16 + r] = c2[v];
        hps[row + 3 * 16 + r] = c3[v];
    }
}

// ---------------------------------------------------------------------------
// Kernel 5: out = elu(sum of 4 segment partials)
// ---------------------------------------------------------------------------
__global__ void k5_reduce_elu(const float* __restrict__ hp,
                              float* __restrict__ out) {
    int idx = blockIdx.x * blockDim.x + threadIdx.x;
    float v = hp[idx] + hp[idx + HP_SEG] + hp[idx + 2 * HP_SEG] + hp[idx + 3 * HP_SEG];
    out[idx] = (v > 0.f) ? v : (__expf(v) - 1.0f);
}

// ---------------------------------------------------------------------------
extern "C" void kernel_launch(void* const* d_in, const int* in_sizes, int n_in,
                              void* d_out, int out_size, void* d_ws, size_t ws_size,
                              hipStream_t stream) {
    const float* x   = (const float*)d_in[0];   // 8192x256
    const int*   adj = (const int*)d_in[1];     // 8192x8192
    const float* W1  = (const float*)d_in[2];   // 256x64
    const float* att = (const float*)d_in[3];   // 128

    float* ws = (float*)d_ws;
    float* h  = ws + WS_H;
    float* s  = ws + WS_S;
    float* dv = ws + WS_D;
    float* b1 = ws + WS_B1;
    float* b2 = ws + WS_B2;
    float* a1 = ws + WS_A1;
    float* a2 = ws + WS_A2;
    float* hp = ws + WS_HP;
    unsigned int* pk = (unsigned int*)(ws + WS_PK);

    float* out = (float*)d_out;

    // 1) h = x @ W1  (WMMA f32)
    k1_gemm_h<<<N_NODES / (16 * 8), 256, 0, stream>>>(x, W1, h);
    // 2) s, d and column exp-factors
    k2_sd<<<N_NODES / 256, 256, 0, stream>>>(h, att, s, dv, b1, b2);
    // 3) single pass over adj: bit-pack + row max + row Z  -> a1, a2
    k3_rowstats<<<N_NODES, 256, 0, stream>>>(adj, s, dv, b1, b2, pk, a1, a2);
    // 4) h_prime partials = attention^T @ h  (WMMA f32, async LDS staging)
    k4_attn_gemm<<<dim3(64, 4), 256, 0, stream>>>(pk, h, s, dv, b1, b2, a1, a2, hp);
    // 5) reduce segments + elu
    k5_reduce_elu<<<(N_NODES * O_OUT) / 256, 256, 0, stream>>>(hp, out);
}